// BlockDCT2D_59906203845015
// MI455X (gfx1250) — compile-verified
//
#include <hip/hip_runtime.h>

typedef __attribute__((ext_vector_type(2))) float v2f;
typedef __attribute__((ext_vector_type(4))) float v4f;
typedef __attribute__((ext_vector_type(8))) float v8f;

#define H_IN 500
#define W_IN 500
#define NC_TOT 512              // N*C = 8*64
#define NBLK 63                 // 8x8 blocks per padded dim (504/8)
#define NTILE 32                // 16x16 tiles per dim, ceil(63/2)
#define TOTAL_TILES (NC_TOT * NTILE * NTILE)   // 524288
#define LDS_STRIDE 19           // conflict-free padded row stride (floats)

// Orthonormal 8x8 DCT-II matrix, D[k][n] = alpha_k * cos(pi*(2n+1)k/16)
__constant__ float cD[64] = {
  0.35355339059327373f,  0.35355339059327373f,  0.35355339059327373f,  0.35355339059327373f,
  0.35355339059327373f,  0.35355339059327373f,  0.35355339059327373f,  0.35355339059327373f,
  0.49039264020161522f,  0.41573480615127262f,  0.27778511650980114f,  0.09754516100806417f,
 -0.09754516100806417f, -0.27778511650980114f, -0.41573480615127262f, -0.49039264020161522f,
  0.46193976625564337f,  0.19134171618254492f, -0.19134171618254492f, -0.46193976625564337f,
 -0.46193976625564337f, -0.19134171618254492f,  0.19134171618254492f,  0.46193976625564337f,
  0.41573480615127262f, -0.09754516100806417f, -0.49039264020161522f, -0.27778511650980114f,
  0.27778511650980114f,  0.49039264020161522f,  0.09754516100806417f, -0.41573480615127262f,
  0.35355339059327373f, -0.35355339059327373f, -0.35355339059327373f,  0.35355339059327373f,
  0.35355339059327373f, -0.35355339059327373f, -0.35355339059327373f,  0.35355339059327373f,
  0.27778511650980114f, -0.49039264020161522f,  0.09754516100806417f,  0.41573480615127262f,
 -0.41573480615127262f, -0.09754516100806417f,  0.49039264020161522f, -0.27778511650980114f,
  0.19134171618254492f, -0.46193976625564337f,  0.46193976625564337f, -0.19134171618254492f,
 -0.19134171618254492f,  0.46193976625564337f, -0.46193976625564337f,  0.19134171618254492f,
  0.09754516100806417f, -0.27778511650980114f,  0.41573480615127262f, -0.49039264020161522f,
  0.49039264020161522f, -0.41573480615127262f,  0.27778511650980114f, -0.09754516100806417f
};

__global__ __launch_bounds__(256) void BlockDCT2D_wmma_kernel(
    const float* __restrict__ x, float* __restrict__ out) {
  __shared__ float lds[8][16 * LDS_STRIDE];

  const int tid  = threadIdx.x;
  const int lane = tid & 31;
  const int wv   = tid >> 5;
  const int hi   = lane >> 4;    // which half-wave
  const int ln   = lane & 15;
  float* myLds = lds[wv];

  // A operand = block-diagonal diag(D, D), in 16x4-slice A layout:
  // lane (m = ln, half hi), vgpr v holds A(m, k') with k' = 2*hi + v of chunk kk.
  v2f A[4];
#pragma unroll
  for (int kk = 0; kk < 4; ++kk) {
#pragma unroll
    for (int v = 0; v < 2; ++v) {
      const int K = 4 * kk + 2 * hi + v;
      float val = 0.0f;
      if ((ln < 8) == (K < 8)) val = cD[(ln & 7) * 8 + (K & 7)];
      A[kk][v] = val;
    }
  }

  const int nwaves = (int)((gridDim.x * blockDim.x) >> 5);
  for (int t = (int)((blockIdx.x * blockDim.x + tid) >> 5); t < TOTAL_TILES; t += nwaves) {
    const int nc  = t >> 10;           // / (32*32)
    const int rem = t & 1023;
    const int ti  = rem >> 5;
    const int tj  = rem & 31;
    const float* src = x + (size_t)nc * (H_IN * W_IN);

    // reflect-pad column index (valid for col <= 511)
    const int col = 16 * tj + ln;
    const int mc  = (col < W_IN) ? col : (2 * W_IN - 2 - col);

    // B operand = X tile in 4x16-slice B layout: B(k', n) at lane n + 16*(k'/2), vgpr k'%2
    v2f B[4];
#pragma unroll
    for (int kk = 0; kk < 4; ++kk) {
#pragma unroll
      for (int v = 0; v < 2; ++v) {
        const int row = 16 * ti + 4 * kk + 2 * hi + v;
        const int mr  = (row < H_IN) ? row : (2 * H_IN - 2 - row);
        B[kk][v] = src[mr * W_IN + mc];
      }
    }

    // U = D2 * X   (K=16 via 4 chained 16x16x4 f32 WMMAs)
    v8f U = {0.f, 0.f, 0.f, 0.f, 0.f, 0.f, 0.f, 0.f};
#pragma unroll
    for (int kk = 0; kk < 4; ++kk)
      U = __builtin_amdgcn_wmma_f32_16x16x4_f32(false, A[kk], false, B[kk],
                                                (short)0, U, false, false);

    // Stage U through LDS (row-major, stride 19 -> bank-conflict free both ways).
    // C layout: reg r of lane holds U(m = r + 8*hi, n = ln).
#pragma unroll
    for (int r = 0; r < 8; ++r)
      myLds[(r + 8 * hi) * LDS_STRIDE + ln] = U[r];

    asm volatile("" ::: "memory");  // compiler fence; DS pipe is in-order per wave

    // B2 = U^T in B layout: need U(ln, 4kk + 2*hi + v)
    v2f B2[4];
#pragma unroll
    for (int kk = 0; kk < 4; ++kk) {
#pragma unroll
      for (int v = 0; v < 2; ++v)
        B2[kk][v] = myLds[ln * LDS_STRIDE + 4 * kk + 2 * hi + v];
    }

    // V = D2 * U^T = (D2 * X * D2^T)^T = Z^T
    v8f V = {0.f, 0.f, 0.f, 0.f, 0.f, 0.f, 0.f, 0.f};
#pragma unroll
    for (int kk = 0; kk < 4; ++kk)
      V = __builtin_amdgcn_wmma_f32_16x16x4_f32(false, A[kk], false, B2[kk],
                                                (short)0, V, false, false);

    // Reg r of lane holds V(m = r + 8*hi, n = ln) = Z(ln, r + 8*hi):
    //   block (bi = 2ti + ln/8, bj = 2tj + hi), coeff row a = ln&7, coeff col e = r.
    // => each lane's 8 regs are 8 consecutive output floats. Fully coalesced b128 stores.
    const int bi = 2 * ti + (ln >> 3);
    const int bj = 2 * tj + hi;
    if (bi < NBLK && bj < NBLK) {
      float* dst = out + (((size_t)nc * NBLK * NBLK) + (size_t)bi * NBLK + (size_t)bj) * 64
                       + (size_t)(ln & 7) * 8;
      v4f lo = {V[0], V[1], V[2], V[3]};
      v4f hh = {V[4], V[5], V[6], V[7]};
      *(v4f*)(dst)     = lo;
      *(v4f*)(dst + 4) = hh;
    }
  }
}

extern "C" void kernel_launch(void* const* d_in, const int* in_sizes, int n_in,
                              void* d_out, int out_size, void* d_ws, size_t ws_size,
                              hipStream_t stream) {
  (void)in_sizes; (void)n_in; (void)out_size; (void)d_ws; (void)ws_size;
  const float* x = (const float*)d_in[0];
  float* out = (float*)d_out;
  // 4096 blocks * 8 waves = 32768 waves, grid-stride over 524288 tiles (16 tiles/wave)
  BlockDCT2D_wmma_kernel<<<dim3(4096), dim3(256), 0, stream>>>(x, out);
}